// StructuralEncoder_11768210391328
// MI455X (gfx1250) — compile-verified
//
#include <hip/hip_runtime.h>
#include <math.h>

typedef __attribute__((ext_vector_type(2))) float v2f;
typedef __attribute__((ext_vector_type(8))) float v8f;

#if defined(__HIP_DEVICE_COMPILE__)
#if !__has_builtin(__builtin_amdgcn_wmma_f32_16x16x4_f32)
#error "wmma_f32_16x16x4_f32 builtin not available on this toolchain (device pass)"
#endif
#endif

#define NHEAD 8
#define HC 12
#define DM 96
#define NEG_SLOPE 0.2f

// ---------------------------------------------------------------------------
// GEMM: H[N,96] = X[N,K] @ W[K,96], fp32 WMMA 16x16x4.
// One wave (32 threads) computes one 16x16 output tile.
// A layout (f32 16x4): lanes 0-15 hold row M=lane, K = {0,1}; lanes 16-31 K = {2,3}.
// B layout (f32 4x16): lanes 0-15 col N=lane, rows {K, K+1}; lanes 16-31 rows {K+2, K+3}.
// C/D layout: VGPR r -> row m0 + r + (lane>=16 ? 8 : 0), col n0 + (lane & 15).
// ---------------------------------------------------------------------------
__global__ void __launch_bounds__(32)
gat_gemm_wmma(const float* __restrict__ X, const float* __restrict__ W,
              float* __restrict__ H, int K) {
    const int m0   = blockIdx.x * 16;
    const int n0   = blockIdx.y * 16;
    const int lane = threadIdx.x;        // 0..31, full wave, no divergence
    const int half = lane >> 4;          // 0 or 1
    const int lr   = lane & 15;

    v8f acc = {};
    const float* xrow = X + (size_t)(m0 + lr) * K + half * 2;
    const float* wcol = W + (n0 + lr) + (size_t)(half * 2) * DM;

    #pragma unroll 4
    for (int k = 0; k < K; k += 4) {
        v2f a = *(const v2f*)(xrow + k);                 // x[m0+lr][k+half*2 .. +1]
        v2f b;
        b.x = wcol[(size_t)k * DM];                      // W[k+half*2    ][n0+lr]
        b.y = wcol[(size_t)k * DM + DM];                 // W[k+half*2 + 1][n0+lr]
        acc = __builtin_amdgcn_wmma_f32_16x16x4_f32(
            false, a, false, b, (short)0, acc, false, false);
    }

    float* hout = H + (size_t)(m0 + half * 8) * DM + n0 + lr;
    #pragma unroll
    for (int r = 0; r < 8; ++r) hout[(size_t)r * DM] = acc[r];
}

// ---------------------------------------------------------------------------
// Per (node, head): a_src/a_dst dot products against attention vectors.
// ---------------------------------------------------------------------------
__global__ void gat_att(const float* __restrict__ H,
                        const float* __restrict__ att_src,
                        const float* __restrict__ att_dst,
                        float* __restrict__ aS, float* __restrict__ aD, int N) {
    int idx = blockIdx.x * blockDim.x + threadIdx.x;
    if (idx >= N * NHEAD) return;
    int n = idx >> 3, hd = idx & 7;
    const float* hp = H + (size_t)n * DM + hd * HC;
    const float* as = att_src + hd * HC;
    const float* ad = att_dst + hd * HC;
    float s = 0.f, d = 0.f;
    #pragma unroll
    for (int c = 0; c < HC; ++c) { float v = hp[c]; s += v * as[c]; d += v * ad[c]; }
    aS[idx] = s; aD[idx] = d;
}

// ---------------------------------------------------------------------------
// Reset segment buffers: emax=-inf, denom=0 (N*8), acc=0 (N*96).
// ---------------------------------------------------------------------------
__global__ void gat_init(float* __restrict__ emax, float* __restrict__ denom,
                         float* __restrict__ acc, int n8, int n96) {
    int i = blockIdx.x * blockDim.x + threadIdx.x;
    if (i < n8) { emax[i] = -INFINITY; denom[i] = 0.f; }
    if (i < n96) acc[i] = 0.f;
}

// float atomic max via monotone int punning (valid with -inf init)
__device__ __forceinline__ void atomicMaxF(float* addr, float val) {
    int bits = __float_as_int(val);
    if (bits >= 0) atomicMax((int*)addr, bits);
    else           atomicMin((unsigned int*)addr, (unsigned int)bits);
}

__device__ __forceinline__ float leaky(float v) {
    return v > 0.f ? v : NEG_SLOPE * v;
}

__device__ __forceinline__ void edge_nodes(const long long* __restrict__ ei,
                                           int e, int E, int& s, int& d) {
    if (e < E) { s = (int)ei[e]; d = (int)ei[E + e]; }
    else       { s = d = e - E; }                      // appended self-loops
}

// ---------------------------------------------------------------------------
// Pass 1: segment max of leaky_relu(a_src[s] + a_dst[d]) into emax[dst].
// ---------------------------------------------------------------------------
__global__ void gat_edge_max(const long long* __restrict__ ei, int E, int ET,
                             const float* __restrict__ aS,
                             const float* __restrict__ aD,
                             float* __restrict__ emax) {
    int e = blockIdx.x * blockDim.x + threadIdx.x;
    if (e >= ET) return;
    int s, d; edge_nodes(ei, e, E, s, d);
    const float4 s0 = *(const float4*)(aS + (size_t)s * 8);
    const float4 s1 = *(const float4*)(aS + (size_t)s * 8 + 4);
    const float4 d0 = *(const float4*)(aD + (size_t)d * 8);
    const float4 d1 = *(const float4*)(aD + (size_t)d * 8 + 4);
    float* em = emax + (size_t)d * 8;
    atomicMaxF(em + 0, leaky(s0.x + d0.x));
    atomicMaxF(em + 1, leaky(s0.y + d0.y));
    atomicMaxF(em + 2, leaky(s0.z + d0.z));
    atomicMaxF(em + 3, leaky(s0.w + d0.w));
    atomicMaxF(em + 4, leaky(s1.x + d1.x));
    atomicMaxF(em + 5, leaky(s1.y + d1.y));
    atomicMaxF(em + 6, leaky(s1.z + d1.z));
    atomicMaxF(em + 7, leaky(s1.w + d1.w));
}

__device__ __forceinline__ void atomicAddF(float* addr, float v) {
    __hip_atomic_fetch_add(addr, v, __ATOMIC_RELAXED, __HIP_MEMORY_SCOPE_AGENT);
}

// ---------------------------------------------------------------------------
// Pass 2: denom[dst] += exp(e - emax[dst]).
// ---------------------------------------------------------------------------
__global__ void gat_edge_sum(const long long* __restrict__ ei, int E, int ET,
                             const float* __restrict__ aS,
                             const float* __restrict__ aD,
                             const float* __restrict__ emax,
                             float* __restrict__ denom) {
    int e = blockIdx.x * blockDim.x + threadIdx.x;
    if (e >= ET) return;
    int s, d; edge_nodes(ei, e, E, s, d);
    const float* as = aS + (size_t)s * 8;
    const float* ad = aD + (size_t)d * 8;
    const float* em = emax + (size_t)d * 8;
    float* dn = denom + (size_t)d * 8;
    #pragma unroll
    for (int h = 0; h < NHEAD; ++h) {
        float v = leaky(as[h] + ad[h]);
        atomicAddF(dn + h, __expf(v - em[h]));
    }
}

// ---------------------------------------------------------------------------
// Pass 3: acc[dst] += alpha * h[src], parallel over (edge, head).
// ---------------------------------------------------------------------------
__global__ void gat_edge_agg(const long long* __restrict__ ei, int E, int ET,
                             const float* __restrict__ aS,
                             const float* __restrict__ aD,
                             const float* __restrict__ emax,
                             const float* __restrict__ denom,
                             const float* __restrict__ H,
                             float* __restrict__ acc) {
    int idx = blockIdx.x * blockDim.x + threadIdx.x;
    if (idx >= ET * NHEAD) return;
    int e = idx >> 3, hd = idx & 7;
    int s, d; edge_nodes(ei, e, E, s, d);
    float v = leaky(aS[(size_t)s * 8 + hd] + aD[(size_t)d * 8 + hd]);
    float alpha = __expf(v - emax[(size_t)d * 8 + hd]) /
                  (denom[(size_t)d * 8 + hd] + 1e-16f);
    const float* hs = H + (size_t)s * DM + hd * HC;
    float* ap = acc + (size_t)d * DM + hd * HC;
    #pragma unroll
    for (int c = 0; c < HC; ++c) atomicAddF(ap + c, hs[c] * alpha);
}

// ---------------------------------------------------------------------------
// out = relu(acc + bias)
// ---------------------------------------------------------------------------
__global__ void gat_finish(const float* __restrict__ acc,
                           const float* __restrict__ bias,
                           float* __restrict__ out, int n96) {
    int i = blockIdx.x * blockDim.x + threadIdx.x;
    if (i >= n96) return;
    float v = acc[i] + bias[i % DM];
    out[i] = v > 0.f ? v : 0.f;
}

// ---------------------------------------------------------------------------
extern "C" void kernel_launch(void* const* d_in, const int* in_sizes, int n_in,
                              void* d_out, int out_size, void* d_ws, size_t ws_size,
                              hipStream_t stream) {
    const float*     x   = (const float*)d_in[0];
    const long long* ei  = (const long long*)d_in[1];   // int64 edge_index [2,E]
    const float*     W1  = (const float*)d_in[2];
    const float*     as1 = (const float*)d_in[3];
    const float*     ad1 = (const float*)d_in[4];
    const float*     b1  = (const float*)d_in[5];
    const float*     W2  = (const float*)d_in[6];
    const float*     as2 = (const float*)d_in[7];
    const float*     ad2 = (const float*)d_in[8];
    const float*     b2  = (const float*)d_in[9];

    const int IN_DIM = 128;
    const int N  = in_sizes[0] / IN_DIM;   // 50000
    const int E  = in_sizes[1] / 2;        // 800000
    const int ET = E + N;                  // + self-loops
    const int N8 = N * NHEAD, N96 = N * DM;

    float* ws     = (float*)d_ws;
    float* bufH   = ws;                    // [N,96] gemm output (per layer)
    float* bufAcc = bufH   + N96;          // [N,96] aggregation accumulator
    float* bufX2  = bufAcc + N96;          // [N,96] layer-2 input
    float* aS     = bufX2  + N96;          // [N,8]
    float* aD     = aS + N8;
    float* emax   = aD + N8;
    float* denom  = emax + N8;

    const int BT = 256;
    dim3 gGemm(N / 16, DM / 16);
    int gAtt  = (N8 + BT - 1) / BT;
    int gInit = (N96 + BT - 1) / BT;
    int gEdge = (ET + BT - 1) / BT;
    int gAgg  = (ET * NHEAD + BT - 1) / BT;
    int gFin  = (N96 + BT - 1) / BT;

    // ---- layer 1 ----
    gat_gemm_wmma<<<gGemm, 32, 0, stream>>>(x, W1, bufH, IN_DIM);
    gat_att<<<gAtt, BT, 0, stream>>>(bufH, as1, ad1, aS, aD, N);
    gat_init<<<gInit, BT, 0, stream>>>(emax, denom, bufAcc, N8, N96);
    gat_edge_max<<<gEdge, BT, 0, stream>>>(ei, E, ET, aS, aD, emax);
    gat_edge_sum<<<gEdge, BT, 0, stream>>>(ei, E, ET, aS, aD, emax, denom);
    gat_edge_agg<<<gAgg, BT, 0, stream>>>(ei, E, ET, aS, aD, emax, denom, bufH, bufAcc);
    gat_finish<<<gFin, BT, 0, stream>>>(bufAcc, b1, bufX2, N96);

    // ---- layer 2 ----
    gat_gemm_wmma<<<gGemm, 32, 0, stream>>>(bufX2, W2, bufH, DM);
    gat_att<<<gAtt, BT, 0, stream>>>(bufH, as2, ad2, aS, aD, N);
    gat_init<<<gInit, BT, 0, stream>>>(emax, denom, bufAcc, N8, N96);
    gat_edge_max<<<gEdge, BT, 0, stream>>>(ei, E, ET, aS, aD, emax);
    gat_edge_sum<<<gEdge, BT, 0, stream>>>(ei, E, ET, aS, aD, emax, denom);
    gat_edge_agg<<<gAgg, BT, 0, stream>>>(ei, E, ET, aS, aD, emax, denom, bufH, bufAcc);
    gat_finish<<<gFin, BT, 0, stream>>>(bufAcc, b2, (float*)d_out, N96);
}